// SGCLayer_12584254177709
// MI455X (gfx1250) — compile-verified
//
#include <hip/hip_runtime.h>
#include <hip/hip_bf16.h>

typedef __attribute__((ext_vector_type(2))) float v2f;
typedef __attribute__((ext_vector_type(8))) float v8f;

#define NNODES 50000
#define DFEAT  64
#define NEDGES 800000

// ---------------------------------------------------------------------------
// zero scratch (float4-wide stores)
// ---------------------------------------------------------------------------
__global__ void sgc_zero_kernel(float4* __restrict__ p, int n4) {
    int i = blockIdx.x * blockDim.x + threadIdx.x;
    if (i < n4) p[i] = make_float4(0.f, 0.f, 0.f, 0.f);
}

// ---------------------------------------------------------------------------
// SpMM scatter: 16 lanes per edge, float4 per lane (256B coalesced tile/edge).
// Gathers hit L2 (x fits in 192MB L2); scatter uses no-return f32 atomics.
// ---------------------------------------------------------------------------
__global__ void sgc_spmm_kernel(const float* __restrict__ xin,
                                float* __restrict__ yout,
                                const long long* __restrict__ src,
                                const long long* __restrict__ dst,
                                const float* __restrict__ attr,
                                int nedges) {
    int gid = blockIdx.x * blockDim.x + threadIdx.x;
    int e = gid >> 4;              // 16 lanes per edge
    int f = (gid & 15) << 2;       // 4 floats per lane
    if (e >= nedges) return;

    long long s = src[e];
    long long d = dst[e];
    float w = attr[e];

    const float4 v = *reinterpret_cast<const float4*>(xin + (size_t)s * DFEAT + f);
    float* o = yout + (size_t)d * DFEAT + f;
    // no-return global_atomic_add_f32 (STOREcnt path, no VGPR writeback)
    unsafeAtomicAdd(o + 0, w * v.x);
    unsafeAtomicAdd(o + 1, w * v.y);
    unsafeAtomicAdd(o + 2, w * v.z);
    unsafeAtomicAdd(o + 3, w * v.w);
}

// ---------------------------------------------------------------------------
// out = Y @ W + b using V_WMMA_F32_16X16X4_F32 (exact fp32 tensor path).
// One wave computes exactly ONE 16x64 output tile (no outer loop -> nothing
// for LLVM to hoist/spill). K loop kept rolled: per iteration one
// global_load_b64 (A), LDS reads for B, four WMMAs. Bias is splatted into
// the accumulators as the initial C value (bias depends only on the column,
// and each acc vector holds a single column per lane), so the epilogue is
// pure stores.
// ---------------------------------------------------------------------------
__global__ void sgc_gemm_wmma_kernel(const float* __restrict__ Y,
                                     const float* __restrict__ W,
                                     const float* __restrict__ bias,
                                     float* __restrict__ out,
                                     int nrows) {
    __shared__ float sW[DFEAT * DFEAT];
    __shared__ float sB[DFEAT];

    for (int i = threadIdx.x; i < DFEAT * DFEAT; i += blockDim.x) sW[i] = W[i];
    if (threadIdx.x < DFEAT) sB[threadIdx.x] = bias[threadIdx.x];
    __syncthreads();

    const int lane = threadIdx.x & 31;
    const int wave = threadIdx.x >> 5;
    const int tile = blockIdx.x * (blockDim.x >> 5) + wave;  // wave-uniform
    if (tile * 16 >= nrows) return;   // uniform branch: EXEC all-ones inside

    const int m  = lane & 15;   // A row / C col within 16
    const int hi = lane >> 4;   // lane half selects K/M offsets
    const int row0 = tile * 16;

    // Seed accumulators with the bias for this lane's output column.
    const float bv0 = sB[m +  0];
    const float bv1 = sB[m + 16];
    const float bv2 = sB[m + 32];
    const float bv3 = sB[m + 48];
    v8f acc0, acc1, acc2, acc3;
    #pragma unroll
    for (int r = 0; r < 8; ++r) {
        acc0[r] = bv0; acc1[r] = bv1; acc2[r] = bv2; acc3[r] = bv3;
    }

    // A operand pointer: row (row0+m), starting K col = 2*hi
    const float* aptr = Y + (size_t)(row0 + m) * DFEAT + 2 * hi;
    // B operand LDS offset: row (2*hi), col m  (float index)
    int lbase = 2 * hi * DFEAT + m;

    #pragma unroll 1
    for (int kk = 0; kk < 16; ++kk) {
        // A 16x4 fp32 tile: VGPR v = A[m][kk*4 + 2*hi + v]  (GLOBAL_LOAD_B64)
        const v2f a = *reinterpret_cast<const v2f*>(aptr);
        aptr += 4;

        v2f b0; b0.x = sW[lbase +  0];          b0.y = sW[lbase + DFEAT +  0];
        v2f b1; b1.x = sW[lbase + 16];          b1.y = sW[lbase + DFEAT + 16];
        v2f b2; b2.x = sW[lbase + 32];          b2.y = sW[lbase + DFEAT + 32];
        v2f b3; b3.x = sW[lbase + 48];          b3.y = sW[lbase + DFEAT + 48];
        lbase += 4 * DFEAT;

        acc0 = __builtin_amdgcn_wmma_f32_16x16x4_f32(false, a, false, b0, (short)0, acc0, false, false);
        acc1 = __builtin_amdgcn_wmma_f32_16x16x4_f32(false, a, false, b1, (short)0, acc1, false, false);
        acc2 = __builtin_amdgcn_wmma_f32_16x16x4_f32(false, a, false, b2, (short)0, acc2, false, false);
        acc3 = __builtin_amdgcn_wmma_f32_16x16x4_f32(false, a, false, b3, (short)0, acc3, false, false);
    }

    // C/D layout: VGPR r -> row (r + 8*hi), col = lane&15. Pure stores.
    v8f accs[4] = {acc0, acc1, acc2, acc3};
    #pragma unroll
    for (int nt = 0; nt < 4; ++nt) {
        const int nn = nt * 16 + m;
        #pragma unroll
        for (int r = 0; r < 8; ++r) {
            const int mm = r + hi * 8;
            out[(size_t)(row0 + mm) * DFEAT + nn] = accs[nt][r];
        }
    }
}

// ---------------------------------------------------------------------------
extern "C" void kernel_launch(void* const* d_in, const int* in_sizes, int n_in,
                              void* d_out, int out_size, void* d_ws, size_t ws_size,
                              hipStream_t stream) {
    const float*      x    = (const float*)d_in[0];
    const long long*  ei   = (const long long*)d_in[1];   // int64 [2, E]
    const float*      attr = (const float*)d_in[2];
    const float*      W    = (const float*)d_in[3];
    const float*      b    = (const float*)d_in[4];
    float*            out  = (float*)d_out;

    const int N = NNODES, E = NEDGES;
    float* y0 = (float*)d_ws;
    float* y1 = y0 + (size_t)N * DFEAT;

    // zero both scatter targets (ws is poisoned; must re-zero every call)
    const int n4 = 2 * N * DFEAT / 4;
    sgc_zero_kernel<<<(n4 + 255) / 256, 256, 0, stream>>>((float4*)y0, n4);

    // two SpMM-scatter rounds
    const int spmmThreads = E * 16;
    const int spmmBlocks  = (spmmThreads + 255) / 256;
    sgc_spmm_kernel<<<spmmBlocks, 256, 0, stream>>>(x,  y0, ei, ei + E, attr, E);
    sgc_spmm_kernel<<<spmmBlocks, 256, 0, stream>>>(y0, y1, ei, ei + E, attr, E);

    // fp32 WMMA GEMM: 3125 row-tiles of 16, one tile per wave, 8 waves/block
    const int tiles  = (N + 15) / 16;
    const int blocks = (tiles + 7) / 8;
    sgc_gemm_wmma_kernel<<<blocks, 256, 0, stream>>>(y1, W, b, out, N);
}